// dDTW_32177894982221
// MI455X (gfx1250) — compile-verified
//
#include <hip/hip_runtime.h>
#include <hip/hip_bf16.h>

typedef __attribute__((ext_vector_type(2))) float v2f;
typedef __attribute__((ext_vector_type(8))) float v8f;

#define B_ 64
#define N_ 512
#define M_ 512
#define D_ 128
#define GAMMA_ 1.0f
#define BIG_ 1e30f

// ---------------------------------------------------------------------------
// Kernel 1: row squared-norms for X and Y.
// t in [0, B*N) -> x2, t in [B*N, 2*B*N) -> y2.
// ---------------------------------------------------------------------------
__global__ void dtw_norms_kernel(const float* __restrict__ X,
                                 const float* __restrict__ Y,
                                 float* __restrict__ x2,
                                 float* __restrict__ y2) {
    int t = blockIdx.x * blockDim.x + threadIdx.x;
    const float* src;
    float* dst;
    int idx;
    if (t < B_ * N_) {
        src = X; dst = x2; idx = t;
    } else {
        src = Y; dst = y2; idx = t - B_ * N_;
    }
    const float4* row = (const float4*)(src + (size_t)idx * D_);
    float s = 0.0f;
#pragma unroll
    for (int k = 0; k < D_ / 4; ++k) {
        float4 v = row[k];
        s += v.x * v.x + v.y * v.y + v.z * v.z + v.w * v.w;
    }
    dst[idx] = s;
}

// ---------------------------------------------------------------------------
// Kernel 2: C[b,i,j] = (x2[i] + y2[j] - 2 * x_i . y_j) / D via f32 WMMA.
// Each wave computes one 16x16 tile of X*Y^T, accumulating 32 WMMAs over K=128.
// A-matrix 16x4 f32 layout (ISA 7.12.2): lane m=L&15, half=L>>4;
//   VGPR0 = A[m][k0+2*half], VGPR1 = A[m][k0+2*half+1]  (contiguous pair -> b64 load)
// B-matrix 4x16 (= Y^T): same per-lane addressing but from Y's row n=L&15.
// C/D layout: VGPR r, lane L -> M = r + 8*(L>>4), N = L&15.
// Block = 256 threads = 8 waves -> 8 column-tiles; grid = (M/128, N/16, B).
// ---------------------------------------------------------------------------
__global__ void dtw_cost_wmma_kernel(const float* __restrict__ X,
                                     const float* __restrict__ Y,
                                     const float* __restrict__ x2,
                                     const float* __restrict__ y2,
                                     float* __restrict__ C) {
    const int lane = threadIdx.x & 31;
    const int wave = threadIdx.x >> 5;
    const int b    = blockIdx.z;
    const int ti16 = blockIdx.y * 16;                   // tile row base (i)
    const int tj16 = (blockIdx.x * 8 + wave) * 16;      // tile col base (j)

    const int m    = lane & 15;
    const int half = lane >> 4;

    const float* xrow = X + ((size_t)(b * N_ + ti16 + m)) * D_;
    const float* yrow = Y + ((size_t)(b * M_ + tj16 + m)) * D_;

    v8f acc = {};
#pragma unroll
    for (int kk = 0; kk < D_ / 4; ++kk) {
        const int k0 = kk * 4 + half * 2;
        v2f a = *(const v2f*)(xrow + k0);
        v2f bb = *(const v2f*)(yrow + k0);
        // D = A(16x4) * B(4x16) + C
        acc = __builtin_amdgcn_wmma_f32_16x16x4_f32(
            /*neg_a=*/false, a, /*neg_b=*/false, bb,
            /*c_mod=*/(short)0, acc, /*reuse_a=*/false, /*reuse_b=*/false);
    }

    const int Nc = tj16 + (lane & 15);
    const float y2v = y2[b * M_ + Nc];
    const float inv_d = 1.0f / (float)D_;
#pragma unroll
    for (int r = 0; r < 8; ++r) {
        const int Mr = ti16 + r + half * 8;
        const float val = (x2[b * N_ + Mr] + y2v - 2.0f * acc[r]) * inv_d;
        C[((size_t)(b * N_ + Mr)) * M_ + Nc] = val;
    }
}

// ---------------------------------------------------------------------------
// Kernel 3: soft-DTW anti-diagonal scan. One block per batch, thread i = row.
// prev  = r on diagonal d-1, prev2 = r on diagonal d-2 (rotated via LDS).
// Cell (i, j=d-i): neighbors (i-1,j)=prev[i-1], (i,j-1)=prev[i], (i-1,j-1)=prev2[i-1].
// ---------------------------------------------------------------------------
__global__ void dtw_scan_kernel(const float* __restrict__ C,
                                float* __restrict__ cost) {
    __shared__ float prev[N_];
    __shared__ float prev2[N_];
    const int i = threadIdx.x;
    const int b = blockIdx.x;
    prev[i]  = BIG_;
    prev2[i] = BIG_;
    __syncthreads();

    const float* Cb = C + (size_t)b * N_ * M_;
    for (int d = 0; d < N_ + M_ - 1; ++d) {
        const int j = d - i;
        const bool valid = (j >= 0) && (j < M_);
        const float c = valid ? Cb[(size_t)i * M_ + j] : 0.0f;

        const float p    = prev[i];
        const float pm1  = (i > 0) ? prev[i - 1]  : BIG_;
        const float p2m1 = (i > 0) ? prev2[i - 1] : BIG_;

        const float a0 = pm1 + c;   // from (i-1, j)
        const float a1 = p + c;     // from (i, j-1)
        const float a2 = p2m1 + c;  // from (i-1, j-1)
        const float mn = fminf(fminf(a0, a1), a2);
        const float s  = __expf((mn - a0) / GAMMA_) +
                         __expf((mn - a1) / GAMMA_) +
                         __expf((mn - a2) / GAMMA_);
        float r = mn - GAMMA_ * __logf(s);
        if (d == 0 && i == 0) r = c;
        if (!valid) r = BIG_;

        __syncthreads();
        prev2[i] = p;
        prev[i]  = r;
        __syncthreads();
    }
    if (i == N_ - 1) cost[b] = prev[N_ - 1];
}

// ---------------------------------------------------------------------------
// Kernel 4: deterministic final reduction: out = sum_b cost[b] / (B*N).
// ---------------------------------------------------------------------------
__global__ void dtw_reduce_kernel(const float* __restrict__ cost,
                                  float* __restrict__ out) {
    if (threadIdx.x == 0 && blockIdx.x == 0) {
        float s = 0.0f;
        for (int b = 0; b < B_; ++b) s += cost[b];
        out[0] = s / ((float)B_ * (float)N_);
    }
}

extern "C" void kernel_launch(void* const* d_in, const int* in_sizes, int n_in,
                              void* d_out, int out_size, void* d_ws, size_t ws_size,
                              hipStream_t stream) {
    (void)in_sizes; (void)n_in; (void)out_size; (void)ws_size;
    const float* X = (const float*)d_in[0];
    const float* Y = (const float*)d_in[1];
    float* out = (float*)d_out;

    // Workspace layout (floats): C | x2 | y2 | per-batch cost
    float* wsC    = (float*)d_ws;
    float* wsX2   = wsC + (size_t)B_ * N_ * M_;
    float* wsY2   = wsX2 + (size_t)B_ * N_;
    float* wsCost = wsY2 + (size_t)B_ * M_;

    dtw_norms_kernel<<<(2 * B_ * N_) / 256, 256, 0, stream>>>(X, Y, wsX2, wsY2);

    dim3 gcost(M_ / 128, N_ / 16, B_);
    dtw_cost_wmma_kernel<<<gcost, 256, 0, stream>>>(X, Y, wsX2, wsY2, wsC);

    dtw_scan_kernel<<<B_, N_, 0, stream>>>(wsC, wsCost);

    dtw_reduce_kernel<<<1, 64, 0, stream>>>(wsCost, out);
}